// PickNMSPredictionsAndReturnAsBatchedResult_16406775071466
// MI455X (gfx1250) — compile-verified
//
#include <hip/hip_runtime.h>
#include <stdint.h>

// Problem constants (from the reference): B=64, N=8192, C=80, M=300, S=19200.
namespace {
constexpr int kB = 64;
constexpr int kN = 8192;
constexpr int kC = 80;
constexpr int kM = 300;
constexpr int kS = kB * kM;            // 19200 selected indexes
constexpr int kChunk = 256;
constexpr int kNChunks = kS / kChunk;  // 75, exact (no tail)
// Flat d_out layout (reference tuple concatenated in return order, as float):
constexpr int kOffNum    = 0;                    // num_predictions (B,1)
constexpr int kOffBoxes  = kB;                   // det_boxes (B,M,4)
constexpr int kOffScores = kB + kB * kM * 4;     // det_scores (B,M)
constexpr int kOffCls    = kB + kB * kM * 5;     // det_classes (B,M)
constexpr int kOutTotal  = kB + kB * kM * 6;     // 115264 floats
constexpr int kOutVec4   = kOutTotal / 4;        // 28816, exact
}  // namespace

// ---- CDNA5 async global->LDS staging (ASYNCcnt path) ----------------------
// Builtin signature (probe-confirmed via hipcc diagnostic):
//   (global int*, shared int*, imm offset, imm cpol)
typedef __attribute__((address_space(1))) int global_int_t;
typedef __attribute__((address_space(3))) int shared_int_t;

#if __has_builtin(__builtin_amdgcn_global_load_async_to_lds_b32)
#define HAVE_ASYNC_LDS 1
#define ASYNC_LOAD_B32(gsrc, ldst)                                            \
  __builtin_amdgcn_global_load_async_to_lds_b32((global_int_t*)(gsrc),        \
                                                (shared_int_t*)(ldst), 0, 0)
#else
#define HAVE_ASYNC_LDS 0
#define ASYNC_LOAD_B32(gsrc, ldst) (*(ldst) = *(gsrc))
#endif

#if __has_builtin(__builtin_amdgcn_s_wait_asynccnt)
#define WAIT_ASYNC0() __builtin_amdgcn_s_wait_asynccnt(0)
#elif HAVE_ASYNC_LDS
#define WAIT_ASYNC0() asm volatile("s_wait_asynccnt 0x0" ::: "memory")
#else
#define WAIT_ASYNC0() ((void)0)
#endif

// ---- Kernel 0: zero the whole output every call (b128 stores) -------------
__global__ __launch_bounds__(256) void nms_zero_kernel(float4* __restrict__ out4) {
  int i = blockIdx.x * 256 + threadIdx.x;
  if (i < kOutVec4) out4[i] = make_float4(0.f, 0.f, 0.f, 0.f);
}

// ---- Kernel 1: per-chunk batch histogram (order-free, LDS atomics) --------
__global__ __launch_bounds__(256) void nms_hist_kernel(const int* __restrict__ sel,
                                                       int* __restrict__ hist) {
  __shared__ int lhist[kB];
  const int t = threadIdx.x;
  const int c = blockIdx.x;
  if (t < kB) lhist[t] = 0;
  __syncthreads();
  const int b = sel[3 * (c * kChunk + t)];  // selected_indexes[i][0]
  atomicAdd(&lhist[b], 1);
  __syncthreads();
  if (t < kB) hist[c * kB + t] = lhist[t];
}

// ---- Kernel 2: exclusive prefix over chunks per batch + num_predictions ---
__global__ __launch_bounds__(64) void nms_scan_kernel(const int* __restrict__ hist,
                                                      int* __restrict__ base,
                                                      float* __restrict__ out) {
  const int b = threadIdx.x;  // one thread per batch, 0..63
  int run = 0;
  for (int c = 0; c < kNChunks; ++c) {
    base[c * kB + b] = run;
    run += hist[c * kB + b];
  }
  out[kOffNum + b] = (float)run;  // num_predictions[b] (total, unclamped)
}

// ---- Kernel 3: stable-rank resolve + gather + scatter ---------------------
__global__ __launch_bounds__(256) void nms_scatter_kernel(
    const float* __restrict__ boxes, const float* __restrict__ scores,
    const int* __restrict__ sel, const int* __restrict__ base,
    float* __restrict__ out) {
  __shared__ int bbuf[kChunk];  // batch ids (de-interleaved during async copy)
  __shared__ int lbuf[kChunk];  // label ids
  __shared__ int xbuf[kChunk];  // box ids
  const int t = threadIdx.x;
  const int c = blockIdx.x;
  const int g = 3 * (c * kChunk + t);

  // Async DMA the chunk's index triples into LDS, de-interleaving via
  // per-lane LDS destination addresses. Tracked with ASYNCcnt.
  ASYNC_LOAD_B32(sel + g + 0, &bbuf[t]);
  ASYNC_LOAD_B32(sel + g + 1, &lbuf[t]);
  ASYNC_LOAD_B32(sel + g + 2, &xbuf[t]);
  WAIT_ASYNC0();
  __syncthreads();

  const int b = bbuf[t];
  // Stable intra-chunk rank: count earlier entries in this chunk with same
  // batch. Uniform loop; bbuf[j] is a broadcast LDS read across all lanes.
  int local_rank = 0;
  for (int j = 0; j < kChunk; ++j) {
    local_rank += (int)((j < t) & (bbuf[j] == b));
  }
  const int rank = base[c * kB + b] + local_rank;

  if (rank < kM) {  // reference drops rank >= M into a discarded overflow row
    const int lab = lbuf[t];
    const int box = xbuf[t];
    const float4 bx =
        *reinterpret_cast<const float4*>(boxes + (size_t)(b * kN + box) * 4);
    const float sc = scores[(size_t)(b * kN + box) * kC + lab];
    // 16B-aligned row -> single global_store_b128
    *reinterpret_cast<float4*>(out + kOffBoxes + (size_t)(b * kM + rank) * 4) = bx;
    out[kOffScores + b * kM + rank] = sc;
    out[kOffCls + b * kM + rank] = (float)lab;  // det_classes as float
  }
}

// ---- Host-side launcher ----------------------------------------------------
extern "C" void kernel_launch(void* const* d_in, const int* in_sizes, int n_in,
                              void* d_out, int out_size, void* d_ws, size_t ws_size,
                              hipStream_t stream) {
  const float* boxes  = (const float*)d_in[0];  // (B,N,4)  f32
  const float* scores = (const float*)d_in[1];  // (B,N,C)  f32
  const int*   sel    = (const int*)d_in[2];    // (S,3)    i32
  float* out = (float*)d_out;

  // Workspace: hist[75][64] then base[75][64]  (38400 bytes total)
  int* hist = (int*)d_ws;
  int* base = hist + kNChunks * kB;

  nms_zero_kernel<<<(kOutVec4 + 255) / 256, 256, 0, stream>>>((float4*)out);
  nms_hist_kernel<<<kNChunks, 256, 0, stream>>>(sel, hist);
  nms_scan_kernel<<<1, 64, 0, stream>>>(hist, base, out);
  nms_scatter_kernel<<<kNChunks, 256, 0, stream>>>(boxes, scores, sel, base, out);
  (void)in_sizes; (void)n_in; (void)out_size; (void)ws_size;
}